// TrajectorySelector_50577534878027
// MI455X (gfx1250) — compile-verified
//
#include <hip/hip_runtime.h>
#include <hip/hip_bf16.h>
#include <math.h>

typedef __attribute__((ext_vector_type(2))) float v2f;
typedef __attribute__((ext_vector_type(8))) float v8f;

constexpr int BB = 32;
constexpr int NN = 256;
constexpr int TT = 64;
constexpr int MM = 64;
constexpr int HH = 512;
constexpr int WW = 512;

// ---------------------------------------------------------------------------
// Kernel 1: one wave32 per (b, n) candidate.
//   f_dis : mean_t ||traj - gt||           (2 timesteps / lane + wave reduce)
//   f_pg  : ||traj[T-1] - goal||
//   f_dac : mean_t (1 - drivable[yi, xi])  (gathers, L2-resident mask)
//   f_col : exp(-(min_{t,m} ||traj_t - obs_m|| - 0.5)^2)
//           min over squared distances computed as 16 x V_WMMA_F32_16X16X4_F32
//           using  d2 = (-2tx)*ox + (-2ty)*oy + |t|^2*1 + 1*|o|^2   (K = 4)
// ---------------------------------------------------------------------------
__global__ __launch_bounds__(256) void score_kernel(
    const float* __restrict__ traj,   // (B,N,T,2)
    const float* __restrict__ goal,   // (B,2)
    const float* __restrict__ gt,     // (B,T,2)
    const float* __restrict__ obst,   // (B,M,2)
    const float* __restrict__ da,     // (B,H,W)
    float* __restrict__ f_dis,
    float* __restrict__ f_pg,
    float* __restrict__ f_col,
    float* __restrict__ f_dac)
{
    const int wave = (blockIdx.x * blockDim.x + threadIdx.x) >> 5;
    const int lane = threadIdx.x & 31;
    const int b  = wave / NN;
    const int bn = wave;

    const float* tp = traj + (size_t)bn * TT * 2;
    const float* gp = gt   + (size_t)b  * TT * 2;
    const float* op = obst + (size_t)b  * MM * 2;
    const float* dp = da   + (size_t)b  * HH * WW;

    // ---- f_dis + f_dac: 2 timesteps per lane ----
    float sdis = 0.f, sdac = 0.f;
#pragma unroll
    for (int r = 0; r < 2; ++r) {
        const int t  = lane + r * 32;
        const float tx = tp[2 * t], ty = tp[2 * t + 1];
        const float dx = tx - gp[2 * t], dy = ty - gp[2 * t + 1];
        sdis += sqrtf(dx * dx + dy * dy);
        int xi = (int)((tx + 50.f) * (0.01f * (float)(WW - 1)));
        int yi = (int)((ty + 50.f) * (0.01f * (float)(HH - 1)));
        xi = min(max(xi, 0), WW - 1);
        yi = min(max(yi, 0), HH - 1);
        sdac += 1.f - dp[yi * WW + xi];
    }

    // ---- f_col via WMMA: build B fragments (obstacles), 4 column tiles ----
    const int  l15 = lane & 15;
    const bool hi  = lane >= 16;
    v2f bf[4];
#pragma unroll
    for (int tm = 0; tm < 4; ++tm) {
        const int j = tm * 16 + l15;
        const float ox = op[2 * j], oy = op[2 * j + 1];
        // lanes 0-15: (K0,K1) = (ox, oy); lanes 16-31: (K2,K3) = (1, |o|^2)
        bf[tm].x = hi ? 1.f : ox;
        bf[tm].y = hi ? (ox * ox + oy * oy) : oy;
    }

    const v8f zeroC = {};
    float mind2 = 3.402823466e38f;
#pragma unroll
    for (int tt = 0; tt < 4; ++tt) {
        const int i = tt * 16 + l15;
        const float tx = tp[2 * i], ty = tp[2 * i + 1];
        v2f af;
        // lanes 0-15: (K0,K1) = (-2tx, -2ty); lanes 16-31: (K2,K3) = (|t|^2, 1)
        af.x = hi ? (tx * tx + ty * ty) : (-2.f * tx);
        af.y = hi ? 1.f : (-2.f * ty);
#pragma unroll
        for (int tm = 0; tm < 4; ++tm) {
            v8f d = __builtin_amdgcn_wmma_f32_16x16x4_f32(
                false, af, false, bf[tm], (short)0, zeroC, false, false);
#pragma unroll
            for (int k = 0; k < 8; ++k) mind2 = fminf(mind2, d[k]);
        }
    }

    // ---- wave32 reductions ----
#pragma unroll
    for (int off = 16; off; off >>= 1) {
        sdis  += __shfl_xor(sdis, off, 32);
        sdac  += __shfl_xor(sdac, off, 32);
        mind2  = fminf(mind2, __shfl_xor(mind2, off, 32));
    }

    // ---- f_pg (endpoint to goal) ----
    const float ex = tp[2 * (TT - 1)], ey = tp[2 * (TT - 1) + 1];
    const float pgx = ex - goal[2 * b], pgy = ey - goal[2 * b + 1];
    const float fpg = sqrtf(pgx * pgx + pgy * pgy);

    if (lane == 0) {
        const float mc = sqrtf(mind2);
        const float e  = mc - 0.5f;
        f_dis[bn] = sdis * (1.f / 64.f);
        f_pg[bn]  = fpg;
        f_col[bn] = expf(-e * e);
        f_dac[bn] = sdac * (1.f / 64.f);
    }
}

// ---------------------------------------------------------------------------
// Kernel 2: one block (256 threads) per batch b.
//   min-max normalize the 4 score arrays over N, score = -(sum),
//   write scores, first-index argmax, gather winning trajectory.
// d_out layout: [0, B*T*2) best trajectories, [B*T*2, B*T*2 + B*N) scores.
// ---------------------------------------------------------------------------
__global__ __launch_bounds__(256) void select_kernel(
    const float* __restrict__ traj,
    const float* __restrict__ f_dis, const float* __restrict__ f_pg,
    const float* __restrict__ f_col, const float* __restrict__ f_dac,
    float* __restrict__ out)
{
    const int b = blockIdx.x;
    const int t = threadIdx.x;   // candidate n

    __shared__ float smn[256];
    __shared__ float smx[256];
    __shared__ float sv[256];
    __shared__ int   si[256];

    float v[4];
    v[0] = f_dis[b * NN + t];
    v[1] = f_pg [b * NN + t];
    v[2] = f_col[b * NN + t];
    v[3] = f_dac[b * NN + t];

    float score = 0.f;
#pragma unroll
    for (int a = 0; a < 4; ++a) {
        smn[t] = v[a];
        smx[t] = v[a];
        __syncthreads();
        for (int s = 128; s; s >>= 1) {
            if (t < s) {
                smn[t] = fminf(smn[t], smn[t + s]);
                smx[t] = fmaxf(smx[t], smx[t + s]);
            }
            __syncthreads();
        }
        const float mn = smn[0];
        const float mx = smx[0];
        __syncthreads();  // allow reuse of smn/smx next round
        float den = mx - mn;
        den = (den == 0.f) ? 1.f : den;
        score += (v[a] - mn) / den;
    }
    score = -score;
    out[BB * TT * 2 + b * NN + t] = score;

    // first-index argmax over N
    sv[t] = score;
    si[t] = t;
    __syncthreads();
    for (int s = 128; s; s >>= 1) {
        if (t < s) {
            const float v2 = sv[t + s];
            const int   i2 = si[t + s];
            if (v2 > sv[t] || (v2 == sv[t] && i2 < si[t])) {
                sv[t] = v2;
                si[t] = i2;
            }
        }
        __syncthreads();
    }
    const int best = si[0];

    if (t < TT * 2) {
        out[b * TT * 2 + t] = traj[((size_t)(b * NN + best) * TT) * 2 + t];
    }
}

extern "C" void kernel_launch(void* const* d_in, const int* in_sizes, int n_in,
                              void* d_out, int out_size, void* d_ws, size_t ws_size,
                              hipStream_t stream) {
    const float* traj = (const float*)d_in[0];
    const float* goal = (const float*)d_in[1];
    const float* gt   = (const float*)d_in[2];
    const float* obst = (const float*)d_in[3];
    const float* da   = (const float*)d_in[4];
    float* out = (float*)d_out;

    float* ws    = (float*)d_ws;        // needs 4 * B*N floats = 128 KB
    float* f_dis = ws;
    float* f_pg  = ws + 1 * BB * NN;
    float* f_col = ws + 2 * BB * NN;
    float* f_dac = ws + 3 * BB * NN;

    // 8192 waves (one per (b,n)), 8 waves per 256-thread block -> 1024 blocks
    score_kernel<<<dim3((BB * NN) / 8), dim3(256), 0, stream>>>(
        traj, goal, gt, obst, da, f_dis, f_pg, f_col, f_dac);

    select_kernel<<<dim3(BB), dim3(256), 0, stream>>>(
        traj, f_dis, f_pg, f_col, f_dac, out);
}